// FocalLoss_69114613728774
// MI455X (gfx1250) — compile-verified
//
#include <hip/hip_runtime.h>
#include <stdint.h>

typedef float f4  __attribute__((ext_vector_type(4)));
typedef float v2f __attribute__((ext_vector_type(2)));
typedef float v8f __attribute__((ext_vector_type(8)));

#define BB 8
#define AA 100000
#define KK 80
#define MM 32
#define NBLK_A ((AA + 255) / 256)   // 391 assignment blocks per image
#define NBLK_B 1024                  // streaming blocks per image
#define F4_PER_IMG (AA * KK / 4)     // 2,000,000 float4s per image

// ---------------------------------------------------------------------------
// Zero the per-image positive-anchor counters (graph replays need re-init).
// ---------------------------------------------------------------------------
__global__ void init_scalars(unsigned* __restrict__ p) {
    int t = threadIdx.x;
    if (t < BB) p[t] = 0u;
}

// ---------------------------------------------------------------------------
// Pass A: anchor<->gt assignment + smooth-L1 loc loss for positives.
// Annotations (MM*5 floats) are staged into LDS with the CDNA5 async
// global->LDS copy path (ASYNCcnt + s_wait_asynccnt).
// ---------------------------------------------------------------------------
__global__ __launch_bounds__(256) void assign_kernel(
    const f4*    __restrict__ anchors,   // [A] (x1,y1,x2,y2)
    const f4*    __restrict__ loc,       // [B*A]
    const float* __restrict__ anno,      // [B*M*5]
    char*        __restrict__ codes,     // [B*A]
    unsigned*    __restrict__ posNum,    // [B]
    float*       __restrict__ locPart)   // [B*NBLK_A]
{
    const int b   = blockIdx.y;
    const int tid = threadIdx.x;

    __shared__ float sAnno[MM * 5];
    if (tid < MM * 5) {
        // LDS byte address = low 32 bits of the generic shared pointer.
        unsigned ldsAddr = (unsigned)(uintptr_t)(&sAnno[tid]);
        unsigned voff    = tid * 4u;
        unsigned long long gbase =
            (unsigned long long)(uintptr_t)(anno + (size_t)b * MM * 5);
        asm volatile("global_load_async_to_lds_b32 %0, %1, %2"
                     :: "v"(ldsAddr), "v"(voff), "s"(gbase) : "memory");
    }
    asm volatile("s_wait_asynccnt 0x0" ::: "memory");
    __syncthreads();

    const int a = blockIdx.x * 256 + tid;
    float locC = 0.0f;
    int   posC = 0;

    if (a < AA) {
        f4 an = anchors[a];
        const float ax1 = an.x, ay1 = an.y, ax2 = an.z, ay2 = an.w;
        const float area_a = (ax2 - ax1) * (ay2 - ay1);

        float best = -2.0f;
        int   arg  = 0;
        #pragma unroll 4
        for (int m = 0; m < MM; ++m) {
            float bx1 = sAnno[m*5+0], by1 = sAnno[m*5+1];
            float bx2 = sAnno[m*5+2], by2 = sAnno[m*5+3];
            float lab = sAnno[m*5+4];
            float area_b = (bx2 - bx1) * (by2 - by1);
            float iw = fmaxf(fminf(ax2, bx2) - fmaxf(ax1, bx1), 0.0f);
            float ih = fmaxf(fminf(ay2, by2) - fmaxf(ay1, by1), 0.0f);
            float inter = iw * ih;
            float uni   = fmaxf(area_a + area_b - inter, 1e-8f);
            float iou   = inter / uni;
            iou = (lab != -1.0f) ? iou : -1.0f;
            if (iou > best) { best = iou; arg = m; }   // first-max, like argmax
        }

        bool pos = best >= 0.5f;
        bool neg = best <  0.4f;
        int  lab = (int)sAnno[arg*5 + 4];
        codes[(size_t)b * AA + a] = pos ? (char)lab : (neg ? (char)-1 : (char)-2);

        if (pos) {
            posC = 1;
            float gx1 = sAnno[arg*5+0], gy1 = sAnno[arg*5+1];
            float gx2 = sAnno[arg*5+2], gy2 = sAnno[arg*5+3];
            float aw  = ax2 - ax1, ah = ay2 - ay1;
            float acx = ax1 + 0.5f * aw, acy = ay1 + 0.5f * ah;
            float gwr = gx2 - gx1, ghr = gy2 - gy1;
            float gcx = gx1 + 0.5f * gwr, gcy = gy1 + 0.5f * ghr;
            float gw  = fmaxf(gwr, 1.0f), gh = fmaxf(ghr, 1.0f);
            f4 pl = loc[(size_t)b * AA + a];
            float d0 = ((gcx - acx) / aw) / 0.1f;
            float d1 = ((gcy - acy) / ah) / 0.1f;
            float d2 = __logf(gw / aw)   / 0.2f;
            float d3 = __logf(gh / ah)   / 0.2f;
            float df[4] = { fabsf(d0 - pl.x), fabsf(d1 - pl.y),
                            fabsf(d2 - pl.z), fabsf(d3 - pl.w) };
            #pragma unroll
            for (int j = 0; j < 4; ++j) {
                float d = df[j];
                locC += (d <= (1.0f/9.0f)) ? 4.5f * d * d : (d - 0.5f/9.0f);
            }
        }
    }

    // wave32 reduce, then block partial (deterministic final sum later)
    #pragma unroll
    for (int off = 16; off > 0; off >>= 1) {
        locC += __shfl_down(locC, off, 32);
        posC += __shfl_down(posC, off, 32);
    }
    __shared__ float wsum[8];
    const int wave = tid >> 5;
    if ((tid & 31) == 0) {
        wsum[wave] = locC;
        if (posC) atomicAdd(&posNum[b], (unsigned)posC);  // integer: deterministic
    }
    __syncthreads();
    if (tid == 0) {
        float s = 0.0f;
        #pragma unroll
        for (int w = 0; w < 8; ++w) s += wsum[w];
        locPart[(size_t)b * NBLK_A + blockIdx.x] = s;
    }
}

// ---------------------------------------------------------------------------
// Pass B: stream 256MB classification with NT b128 loads; focal loss per
// element; block reduction via v_wmma_f32_16x16x4_f32 (A = 64 partials in
// 2 VGPRs, B = ones -> D rows hold the 16 row-sums; 7 adds + shfl_xor(16)
// complete a full 256->1 f32 reduction).
// ---------------------------------------------------------------------------
__global__ __launch_bounds__(256) void cls_kernel(
    const f4*   __restrict__ clsbase,   // [B*A*K/4]
    const char* __restrict__ codes,     // [B*A]
    float*      __restrict__ clsPart)   // [B*NBLK_B]
{
    const int b   = blockIdx.y;
    const int tid = threadIdx.x;
    const f4*   base  = clsbase + (size_t)b * F4_PER_IMG;
    const char* cbase = codes   + (size_t)b * AA;
    const int   stride = gridDim.x * 256;

    float local = 0.0f;
    for (int i = blockIdx.x * 256 + tid; i < F4_PER_IMG; i += stride) {
        __builtin_prefetch((const void*)(base + i + stride), 0, 0);   // global_prefetch_b8
        f4 p4 = __builtin_nontemporal_load(base + i);                  // b128 th:NT
        int anchor = i / (KK / 4);
        int kb     = (i - anchor * (KK / 4)) * 4;
        int code   = (int)cbase[anchor];
        if (code != -2) {           // -2 = ignored anchor -> zero contribution
            #pragma unroll
            for (int j = 0; j < 4; ++j) {
                float p = fminf(fmaxf(p4[j], 1e-4f), 1.0f - 1e-4f);
                float c = (code == kb + j)
                        ? 0.25f * (1.0f - p) * (1.0f - p) * (-__logf(p))
                        : 0.75f * p * p * (-__logf(1.0f - p));
                local += c;
            }
        }
    }

    __shared__ float red[256];
    red[tid] = local;
    __syncthreads();
    if (tid < 32) {   // wave 0 only: EXEC all-ones, WMMA-legal
        float a0 = red[tid]      + red[tid + 64]  + red[tid + 128] + red[tid + 192];
        float a1 = red[tid + 32] + red[tid + 96]  + red[tid + 160] + red[tid + 224];
        v2f Am; Am[0] = a0; Am[1] = a1;          // 16x4 f32 A-matrix (2 VGPRs)
        v2f Bo; Bo[0] = 1.0f; Bo[1] = 1.0f;      // 4x16 all-ones B
        v8f C = {0.0f,0.0f,0.0f,0.0f,0.0f,0.0f,0.0f,0.0f};
        v8f D = __builtin_amdgcn_wmma_f32_16x16x4_f32(
                    false, Am, false, Bo, (short)0, C, false, false);
        float s = D[0]+D[1]+D[2]+D[3]+D[4]+D[5]+D[6]+D[7]; // rowsums 0..7 / 8..15
        s += __shfl_xor(s, 16, 32);                        // join the two halves
        if (tid == 0) clsPart[(size_t)b * NBLK_B + blockIdx.x] = s;
    }
}

// ---------------------------------------------------------------------------
// Pass C: deterministic final sums + per-image normalization + batch mean.
// One wave per image (8 waves).
// ---------------------------------------------------------------------------
__global__ __launch_bounds__(256) void finalize_kernel(
    const float*    __restrict__ anno,
    const unsigned* __restrict__ posNum,
    const float*    __restrict__ locPart,
    const float*    __restrict__ clsPart,
    float*          __restrict__ out)
{
    const int tid  = threadIdx.x;
    const int b    = tid >> 5;
    const int lane = tid & 31;

    float lsum = 0.0f, csum = 0.0f;
    for (int i = lane; i < NBLK_A; i += 32) lsum += locPart[(size_t)b * NBLK_A + i];
    for (int i = lane; i < NBLK_B; i += 32) csum += clsPart[(size_t)b * NBLK_B + i];
    #pragma unroll
    for (int off = 16; off > 0; off >>= 1) {
        lsum += __shfl_down(lsum, off, 32);
        csum += __shfl_down(csum, off, 32);
    }

    __shared__ float sc[BB], sl[BB];
    if (lane == 0) {
        int nv = 0;
        for (int m = 0; m < MM; ++m)
            nv += (anno[((size_t)b * MM + m) * 5 + 4] != -1.0f) ? 1 : 0;
        float pn = (float)posNum[b];
        float c  = csum / fmaxf(pn, 1.0f);
        float l  = (pn > 0.0f) ? (lsum / fmaxf(pn * 4.0f, 1.0f)) : 0.0f;
        sc[b] = (nv > 0) ? c : 0.0f;
        sl[b] = (nv > 0) ? l : 0.0f;
    }
    __syncthreads();
    if (tid == 0) {
        float s = 0.0f, s2 = 0.0f;
        #pragma unroll
        for (int i = 0; i < BB; ++i) { s += sc[i]; s2 += sl[i]; }
        out[0] = s  / (float)BB;
        out[1] = s2 / (float)BB;
    }
}

// ---------------------------------------------------------------------------
extern "C" void kernel_launch(void* const* d_in, const int* in_sizes, int n_in,
                              void* d_out, int out_size, void* d_ws, size_t ws_size,
                              hipStream_t stream) {
    (void)in_sizes; (void)n_in; (void)out_size; (void)ws_size;
    const float* classification = (const float*)d_in[0];   // [B,A,K]
    const float* localization   = (const float*)d_in[1];   // [B,A,4]
    const float* anchors        = (const float*)d_in[2];   // [A,4]
    const float* annotations    = (const float*)d_in[3];   // [B,M,5]
    float* out = (float*)d_out;                            // 2 floats

    // Workspace layout
    char*  codes = (char*)d_ws;                                    // B*A bytes
    size_t off   = (((size_t)BB * AA) + 255) & ~(size_t)255;
    unsigned* posNum = (unsigned*)((char*)d_ws + off);             // [B]
    float* locPart   = (float*)(posNum + BB);                      // [B*NBLK_A]
    float* clsPart   = locPart + (size_t)BB * NBLK_A;              // [B*NBLK_B]

    init_scalars<<<dim3(1), dim3(32), 0, stream>>>(posNum);
    assign_kernel<<<dim3(NBLK_A, BB), dim3(256), 0, stream>>>(
        (const f4*)anchors, (const f4*)localization, annotations,
        codes, posNum, locPart);
    cls_kernel<<<dim3(NBLK_B, BB), dim3(256), 0, stream>>>(
        (const f4*)classification, codes, clsPart);
    finalize_kernel<<<dim3(1), dim3(256), 0, stream>>>(
        annotations, posNum, locPart, clsPart, out);
}